// KDEFunction_83476984365281
// MI455X (gfx1250) — compile-verified
//
#include <hip/hip_runtime.h>
#include <hip/hip_bf16.h>
#include <math.h>

#define B_ROWS   8192
#define N_PTS    16384
#define D_DIM    256
#define BM       128          // rows of x per workgroup (8 waves x 16)
#define BN       32           // X columns per chunk (2 wmma tiles)
#define NP       8            // N-range partitions (grid.y)
#define NSLICE   (N_PTS / NP) // 2048
#define NCHUNK   (NSLICE / BN)// 64
#define LDS_STRIDE 260        // 256 + 4 pad -> conflict-free 16-row frag reads

typedef float v2f __attribute__((ext_vector_type(2)));
typedef float v8f __attribute__((ext_vector_type(8)));

// log_gauss_norm = -(D/2)*log(2*pi*bw^2) = -128*log(0.02*pi)
#define LOG_GAUSS_NORM 354.2135193060795f

// ---------------- kernel 1: colterm[n] = W[n] - 50*||X_n||^2 ----------------
__global__ void kde_colterm_kernel(const float* __restrict__ X,
                                   const float* __restrict__ W,
                                   float* __restrict__ colterm) {
    int n    = (blockIdx.x * blockDim.x + threadIdx.x) >> 5; // one wave32 per row
    int lane = threadIdx.x & 31;
    if (n >= N_PTS) return;
    const float* row = X + (size_t)n * D_DIM;
    float acc = 0.0f;
#pragma unroll
    for (int i = 0; i < D_DIM / 32; ++i) {
        float v = row[lane + i * 32];
        acc = fmaf(v, v, acc);
    }
#pragma unroll
    for (int off = 16; off; off >>= 1) acc += __shfl_xor(acc, off, 32);
    if (lane == 0) colterm[n] = W[n] - 50.0f * acc;
}

// ---------------- kernel 2: ws[0] = logsumexp(W) ----------------
__global__ void kde_lsew_kernel(const float* __restrict__ W, float* __restrict__ ws) {
    __shared__ float red[256];
    int t = threadIdx.x;
    float mx = -INFINITY;
    for (int i = t; i < N_PTS; i += 256) mx = fmaxf(mx, W[i]);
    red[t] = mx; __syncthreads();
    for (int s = 128; s; s >>= 1) { if (t < s) red[t] = fmaxf(red[t], red[t + s]); __syncthreads(); }
    mx = red[0]; __syncthreads();
    float sm = 0.0f;
    for (int i = t; i < N_PTS; i += 256) sm += __expf(W[i] - mx);
    red[t] = sm; __syncthreads();
    for (int s = 128; s; s >>= 1) { if (t < s) red[t] += red[t + s]; __syncthreads(); }
    if (t == 0) ws[0] = mx + __logf(red[0]);
}

// ---------------- kernel 3: streaming GEMM (f32 WMMA) + online logsumexp ----
__launch_bounds__(256, 1)
__global__ void kde_main_kernel(const float* __restrict__ x,
                                const float* __restrict__ X,
                                const float* __restrict__ colterm,
                                float* __restrict__ pm,
                                float* __restrict__ ps) {
    __shared__ float Xs[BN * LDS_STRIDE]; // 33,280 B

    const int tid   = threadIdx.x;
    const int lane  = tid & 31;
    const int w     = tid >> 5;            // wave 0..7 -> owns 16 rows
    const int rb    = blockIdx.x * BM;     // x row base
    const int nbase = blockIdx.y * NSLICE; // this partition's n range
    const int mrow  = lane & 15;           // M (and B-frag N) index
    const int khalf = lane >> 4;           // 0: K{0,1}, 1: K{2,3} per ISA A/B layout

    // ---- preload this wave's A fragments: rows rb+w*16+mrow, all K in regs ----
    v2f afrag[D_DIM / 4];
    {
        const float* xrow = x + (size_t)(rb + w * 16 + mrow) * D_DIM + khalf * 2;
#pragma unroll
        for (int kk = 0; kk < D_DIM / 4; ++kk)
            afrag[kk] = *(const v2f*)(xrow + kk * 4);
    }

    // online logsumexp state per accumulator slot (2 tiles x 8 vgpr rows)
    float m0[8], s0[8], m1[8], s1[8];
#pragma unroll
    for (int r = 0; r < 8; ++r) { m0[r] = -INFINITY; s0[r] = 0.f; m1[r] = -INFINITY; s1[r] = 0.f; }

    for (int c = 0; c < NCHUNK; ++c) {
        const int nb = nbase + c * BN;
        __syncthreads(); // previous chunk's LDS reads done
        // cooperative load X chunk [BN x 256] -> LDS (2048 float4, 8 per thread)
#pragma unroll
        for (int i = 0; i < 8; ++i) {
            int idx = tid + i * 256;
            int row = idx >> 6;
            int cv  = idx & 63;
            float4 v = *(const float4*)(X + (size_t)(nb + row) * D_DIM + cv * 4);
            *(float4*)(&Xs[row * LDS_STRIDE + cv * 4]) = v;
        }
        __syncthreads();

        if (c + 1 < NCHUNK) { // speculative pull of next chunk toward L2/L0
            const float* nxt = X + (size_t)(nb + BN) * D_DIM;
            __builtin_prefetch(nxt + tid * 32, 0, 0);
        }

        v8f acc0 = {0.f, 0.f, 0.f, 0.f, 0.f, 0.f, 0.f, 0.f};
        v8f acc1 = {0.f, 0.f, 0.f, 0.f, 0.f, 0.f, 0.f, 0.f};
        const float* bs0 = &Xs[mrow * LDS_STRIDE + khalf * 2];        // tile0: cols nb..nb+15
        const float* bs1 = &Xs[(16 + mrow) * LDS_STRIDE + khalf * 2]; // tile1: cols nb+16..+31
#pragma unroll
        for (int kk = 0; kk < D_DIM / 4; ++kk) {
            v2f b0 = *(const v2f*)(bs0 + kk * 4);
            v2f b1 = *(const v2f*)(bs1 + kk * 4);
            acc0 = __builtin_amdgcn_wmma_f32_16x16x4_f32(false, afrag[kk], false, b0,
                                                         (short)0, acc0, false, false);
            acc1 = __builtin_amdgcn_wmma_f32_16x16x4_f32(false, afrag[kk], false, b1,
                                                         (short)0, acc1, false, false);
        }

        // score = 100*S + colterm[n]; per-slot online (m,s) update
        float ct0 = colterm[nb + mrow];
        float ct1 = colterm[nb + 16 + mrow];
#pragma unroll
        for (int r = 0; r < 8; ++r) {
            float v  = fmaf(100.0f, acc0[r], ct0);
            float mt = fmaxf(m0[r], v);
            s0[r] = s0[r] * __expf(m0[r] - mt) + __expf(v - mt);
            m0[r] = mt;
            v  = fmaf(100.0f, acc1[r], ct1);
            mt = fmaxf(m1[r], v);
            s1[r] = s1[r] * __expf(m1[r] - mt) + __expf(v - mt);
            m1[r] = mt;
        }
    }

    // merge tile0/tile1 then butterfly across the 16-lane half (xor 1,2,4,8)
#pragma unroll
    for (int r = 0; r < 8; ++r) {
        float m = fmaxf(m0[r], m1[r]);
        float s = s0[r] * __expf(m0[r] - m) + s1[r] * __expf(m1[r] - m);
#pragma unroll
        for (int off = 8; off; off >>= 1) {
            float mo = __shfl_xor(m, off, 32);
            float so = __shfl_xor(s, off, 32);
            float mn = fmaxf(m, mo);
            s = s * __expf(m - mn) + so * __expf(mo - mn);
            m = mn;
        }
        if ((lane & 15) == 0) { // C/D layout: vgpr r -> M=r (lanes 0-15) / M=r+8 (16-31)
            int row = rb + w * 16 + r + khalf * 8;
            pm[blockIdx.y * B_ROWS + row] = m;
            ps[blockIdx.y * B_ROWS + row] = s;
        }
    }
}

// ---------------- kernel 4: merge NP partials, add per-row constant ---------
__global__ void kde_combine_kernel(const float* __restrict__ x,
                                   const float* __restrict__ ws, // [0] = lseW
                                   const float* __restrict__ pm,
                                   const float* __restrict__ ps,
                                   float* __restrict__ out) {
    int b    = (blockIdx.x * blockDim.x + threadIdx.x) >> 5; // one wave32 per row
    int lane = threadIdx.x & 31;
    if (b >= B_ROWS) return;

    const float* xr = x + (size_t)b * D_DIM;
    float acc = 0.0f;
#pragma unroll
    for (int i = 0; i < D_DIM / 32; ++i) {
        float v = xr[lane + i * 32];
        acc = fmaf(v, v, acc);
    }
#pragma unroll
    for (int off = 16; off; off >>= 1) acc += __shfl_xor(acc, off, 32);

    float m = -INFINITY, s = 0.0f;
    if (lane < NP) { m = pm[lane * B_ROWS + b]; s = ps[lane * B_ROWS + b]; }
#pragma unroll
    for (int off = NP / 2; off; off >>= 1) { // xor stays within lanes 0..7 for lane 0
        float mo = __shfl_xor(m, off, 32);
        float so = __shfl_xor(s, off, 32);
        float mn = fmaxf(m, mo);
        s = s * __expf(m - mn) + so * __expf(mo - mn);
        m = mn;
    }
    if (lane == 0)
        out[b] = m + __logf(s) + LOG_GAUSS_NORM - ws[0] - 50.0f * acc;
}

// ---------------- launcher ----------------
extern "C" void kernel_launch(void* const* d_in, const int* in_sizes, int n_in,
                              void* d_out, int out_size, void* d_ws, size_t ws_size,
                              hipStream_t stream) {
    const float* x = (const float*)d_in[0]; // [B, D]
    const float* X = (const float*)d_in[1]; // [N, D]
    const float* W = (const float*)d_in[2]; // [N]
    float* out = (float*)d_out;             // [B]

    float* ws      = (float*)d_ws;
    float* colterm = ws + 64;                    // N_PTS floats
    float* pm      = colterm + N_PTS;            // NP*B floats
    float* ps      = pm + NP * B_ROWS;           // NP*B floats
    (void)in_sizes; (void)n_in; (void)out_size; (void)ws_size;

    kde_lsew_kernel<<<1, 256, 0, stream>>>(W, ws);
    kde_colterm_kernel<<<(N_PTS * 32) / 256, 256, 0, stream>>>(X, W, colterm);

    dim3 grid(B_ROWS / BM, NP);
    kde_main_kernel<<<grid, 256, 0, stream>>>(x, X, colterm, pm, ps);

    kde_combine_kernel<<<(B_ROWS * 32) / 256, 256, 0, stream>>>(x, ws, pm, ps, out);
}